// HeteroSAGE_51711406244224
// MI455X (gfx1250) — compile-verified
//
#include <hip/hip_runtime.h>
#include <stdint.h>

#define DD 128
#define LDA 132   // padded LDS pitch for A tile: bank = (m*132 + k) % 64 -> conflict-free

typedef __attribute__((ext_vector_type(2))) float v2f;
typedef __attribute__((ext_vector_type(8))) float v8f;

// ---------------------------------------------------------------------------
// deg[dst[e]] += 1
__global__ void degree_count(const int* __restrict__ dst, float* __restrict__ deg, int E) {
    int e = blockIdx.x * blockDim.x + threadIdx.x;
    if (e < E) atomicAdd(&deg[dst[e]], 1.0f);
}

// deg[i] = 1 / max(deg[i], 1)
__global__ void reciprocal_max1(float* __restrict__ deg, int n) {
    int i = blockIdx.x * blockDim.x + threadIdx.x;
    if (i < n) deg[i] = 1.0f / fmaxf(deg[i], 1.0f);
}

// agg[dst[e]] += hsrc[src[e]] : one wave32 per edge, float4 per lane (128 feats).
// Edge index is wave-uniform -> readfirstlane so src/dst become scalar loads.
__global__ void scatter_add(const float* __restrict__ hsrc,
                            const int* __restrict__ src,
                            const int* __restrict__ dst,
                            float* __restrict__ agg, int E) {
    int tid = blockIdx.x * blockDim.x + threadIdx.x;
    int e = __builtin_amdgcn_readfirstlane(tid >> 5);
    if (e >= E) return;
    int c = (tid & 31) << 2;
    int s = src[e];
    int d = dst[e];
    float4 v = *(const float4*)(hsrc + (size_t)s * DD + c);
    float* p = agg + (size_t)d * DD + c;
    atomicAdd(p + 0, v.x);
    atomicAdd(p + 1, v.y);
    atomicAdd(p + 2, v.z);
    atomicAdd(p + 3, v.w);
}

// ---------------------------------------------------------------------------
// Fused SAGE layer GEMM (fp32 WMMA):
//   out = hdst @ Ws + (agg * invdeg) @ Wn + bias + hdst   [+ ReLU]
// Block: 256 threads = 8 waves. Block handles a 16-row M strip; wave w handles
// output columns [16w, 16w+16). K=128 stepped by 4 via V_WMMA_F32_16X16X4_F32.
// Both weight matrices are fetched with GLOBAL_LOAD_ASYNC_TO_LDS_B128 into
// double-buffered LDS so the Wn fill overlaps the pass-1 WMMA loop (ASYNCcnt:
// async loads retire in order, so waiting <=16 means the 16 Ws loads landed).
// n (rows) must be a multiple of 16 (true: 50000, 100000).
__global__ __launch_bounds__(256) void sage_gemm(
    const float* __restrict__ hdst,    // [n,128] self features (also residual)
    const float* __restrict__ agg,     // [n,128] neighbor sums
    const float* __restrict__ invdeg,  // [n]     1/max(deg,1)
    const float* __restrict__ Ws,      // [128,128]
    const float* __restrict__ Wn,      // [128,128]
    const float* __restrict__ bias,    // [128]
    float* __restrict__ outp,          // [n,128]
    int relu) {
    __shared__ float lds_ws[DD * DD];   // 64 KB Ws
    __shared__ float lds_wn[DD * DD];   // 64 KB Wn
    __shared__ float lds_a1[16 * LDA];  // 8.25 KB hdst tile
    __shared__ float lds_a2[16 * LDA];  // 8.25 KB mean tile

    const int t     = threadIdx.x;
    const int lane  = t & 31;
    const int wave  = t >> 5;
    const int row0  = blockIdx.x << 4;
    const int n0    = wave << 4;
    const int mm    = lane & 15;             // A row within tile / B,C col within tile
    const int kb    = (lane >> 4) << 1;      // K sub-offset: 0 (lanes 0-15) or 2 (16-31)
    const int half8 = (lane >> 4) << 3;      // C/D row offset: 0 or 8

    // ---- issue async weight DMA: Ws then Wn, 16B/lane/instr, 16 instrs each ----
    {
        unsigned boff = (unsigned)t * 16u;                       // byte offset
        unsigned lws  = (unsigned)(size_t)lds_ws + boff;         // LDS byte addr (low 32 bits)
        unsigned lwn  = (unsigned)(size_t)lds_wn + boff;
        unsigned voff = boff;
        uint64_t gws  = (uint64_t)(uintptr_t)Ws;
        uint64_t gwn  = (uint64_t)(uintptr_t)Wn;
#pragma unroll
        for (int i = 0; i < 16; ++i) {
            asm volatile("global_load_async_to_lds_b128 %0, %1, %2"
                         :: "v"(lws), "v"(voff), "s"(gws) : "memory");
            lws += 4096; voff += 4096;
        }
        voff = boff;
#pragma unroll
        for (int i = 0; i < 16; ++i) {
            asm volatile("global_load_async_to_lds_b128 %0, %1, %2"
                         :: "v"(lwn), "v"(voff), "s"(gwn) : "memory");
            lwn += 4096; voff += 4096;
        }
    }

    // ---- stage both A tiles through VGPRs (needs the invdeg multiply) ----
    for (int i = t; i < 16 * DD / 4; i += 256) {
        int r  = i >> 5;
        int cc = (i & 31) << 2;
        float4 v = *(const float4*)(hdst + (size_t)(row0 + r) * DD + cc);
        *(float4*)(&lds_a1[r * LDA + cc]) = v;
        float sc = invdeg[row0 + r];
        float4 w = *(const float4*)(agg + (size_t)(row0 + r) * DD + cc);
        w.x *= sc; w.y *= sc; w.z *= sc; w.w *= sc;
        *(float4*)(&lds_a2[r * LDA + cc]) = w;
    }

    // Accumulator init = residual + bias (C/D layout: vgpr r -> M=r or r+8, N=lane&15)
    v8f c;
#pragma unroll
    for (int r = 0; r < 8; ++r) {
        int gr = row0 + r + half8;
        int gc = n0 + mm;
        c[r] = hdst[(size_t)gr * DD + gc] + bias[gc];
    }

    // Ws landed (first 16 async ops of this wave retired); barrier makes all
    // waves' Ws portions + A tiles visible. Wn DMA keeps streaming underneath.
    asm volatile("s_wait_asynccnt 16" ::: "memory");
    __syncthreads();

    // ---- pass 1: self term (A = hdst strip, B = Ws) ----
#pragma unroll 4
    for (int k = 0; k < DD; k += 4) {
        v2f a, b;
        a.x = lds_a1[mm * LDA + k + kb];
        a.y = lds_a1[mm * LDA + k + kb + 1];
        b.x = lds_ws[(k + kb) * DD + n0 + mm];
        b.y = lds_ws[(k + kb + 1) * DD + n0 + mm];
        c = __builtin_amdgcn_wmma_f32_16x16x4_f32(false, a, false, b, (short)0, c,
                                                  false, false);
    }

    // Wn landed; barrier covers all waves.
    asm volatile("s_wait_asynccnt 0" ::: "memory");
    __syncthreads();

    // ---- pass 2: neighbor term (A = agg*invdeg strip, B = Wn) ----
#pragma unroll 4
    for (int k = 0; k < DD; k += 4) {
        v2f a, b;
        a.x = lds_a2[mm * LDA + k + kb];
        a.y = lds_a2[mm * LDA + k + kb + 1];
        b.x = lds_wn[(k + kb) * DD + n0 + mm];
        b.y = lds_wn[(k + kb + 1) * DD + n0 + mm];
        c = __builtin_amdgcn_wmma_f32_16x16x4_f32(false, a, false, b, (short)0, c,
                                                  false, false);
    }

    // ---- epilogue ----
#pragma unroll
    for (int r = 0; r < 8; ++r) {
        float v = c[r];
        if (relu) v = fmaxf(v, 0.0f);
        outp[(size_t)(row0 + r + half8) * DD + n0 + mm] = v;
    }
}

// ---------------------------------------------------------------------------
extern "C" void kernel_launch(void* const* d_in, const int* in_sizes, int n_in,
                              void* d_out, int out_size, void* d_ws, size_t ws_size,
                              hipStream_t stream) {
    const float* x_user = (const float*)d_in[0];
    const float* x_item = (const float*)d_in[1];
    const float* Wself  = (const float*)d_in[2];
    const float* Wneigh = (const float*)d_in[3];
    const float* bias   = (const float*)d_in[4];
    const int* src_ui   = (const int*)d_in[5];
    const int* dst_ui   = (const int*)d_in[6];
    const int* src_iu   = (const int*)d_in[7];
    const int* dst_iu   = (const int*)d_in[8];

    const int NU = in_sizes[0] / DD;
    const int NI = in_sizes[1] / DD;
    const int E  = in_sizes[5];

    float* out_u = (float*)d_out;
    float* out_i = out_u + (size_t)NU * DD;

    float* p     = (float*)d_ws;
    float* cur_u = p;  p += (size_t)NU * DD;
    float* cur_i = p;  p += (size_t)NI * DD;
    float* agg_u = p;  p += (size_t)NU * DD;
    float* agg_i = p;  p += (size_t)NI * DD;
    float* inv_u = p;  p += NU;
    float* inv_i = p;  // contiguous with inv_u

    // Degrees are layer-invariant: compute once per call.
    hipMemsetAsync(inv_u, 0, (size_t)(NU + NI) * sizeof(float), stream);
    degree_count<<<(E + 255) / 256, 256, 0, stream>>>(dst_iu, inv_u, E);
    degree_count<<<(E + 255) / 256, 256, 0, stream>>>(dst_ui, inv_i, E);
    reciprocal_max1<<<(NU + NI + 255) / 256, 256, 0, stream>>>(inv_u, NU + NI);

    const float* hu = x_user;
    const float* hi = x_item;
    for (int l = 0; l < 3; ++l) {
        // ping-pong: l0 -> d_out, l1 -> ws, l2 -> d_out (final result lands in d_out)
        float* nu_ = (l == 1) ? cur_u : out_u;
        float* ni_ = (l == 1) ? cur_i : out_i;

        const float* Ws0 = Wself  + (size_t)(l * 2 + 0) * DD * DD;
        const float* Wn0 = Wneigh + (size_t)(l * 2 + 0) * DD * DD;
        const float* b0  = bias   + (size_t)(l * 2 + 0) * DD;
        const float* Ws1 = Wself  + (size_t)(l * 2 + 1) * DD * DD;
        const float* Wn1 = Wneigh + (size_t)(l * 2 + 1) * DD * DD;
        const float* b1  = bias   + (size_t)(l * 2 + 1) * DD;
        int relu = (l < 2) ? 1 : 0;

        // relation 0: user -> item (dst = items)
        hipMemsetAsync(agg_i, 0, (size_t)NI * DD * sizeof(float), stream);
        scatter_add<<<(E * 32 + 255) / 256, 256, 0, stream>>>(hu, src_ui, dst_ui, agg_i, E);
        sage_gemm<<<NI / 16, 256, 0, stream>>>(hi, agg_i, inv_i, Ws0, Wn0, b0, ni_, relu);

        // relation 1: item -> user (dst = users)
        hipMemsetAsync(agg_u, 0, (size_t)NU * DD * sizeof(float), stream);
        scatter_add<<<(E * 32 + 255) / 256, 256, 0, stream>>>(hi, src_iu, dst_iu, agg_u, E);
        sage_gemm<<<NU / 16, 256, 0, stream>>>(hu, agg_u, inv_u, Ws1, Wn1, b1, nu_, relu);

        hu = nu_;
        hi = ni_;
    }
}